// DFineInitialQueryAndReferenceGenerator_56100862821020
// MI455X (gfx1250) — compile-verified
//
#include <hip/hip_runtime.h>
#include <cstdint>

#define B_ 32
#define A_ 8400
#define L_ 80
#define H_ 256
#define K_ 300

// Order-preserving float -> uint map (larger float => larger uint).
__device__ __forceinline__ unsigned f2ord(float f) {
    unsigned u = __float_as_uint(f);
    return (u & 0x80000000u) ? ~u : (u | 0x80000000u);
}

// ---------------------------------------------------------------------------
// Kernel 1: per-row max over L=80, store sortable key. One thread per row.
// ---------------------------------------------------------------------------
__global__ void rowmax_kernel(const float* __restrict__ cls,
                              unsigned* __restrict__ keys) {
    int r = blockIdx.x * blockDim.x + threadIdx.x;   // r in [0, B*A)
    if (r >= B_ * A_) return;
    const float* rowp = cls + (size_t)r * L_;
    // CDNA5 prefetch path (global_prefetch_b8): warm the tail of the row.
    __builtin_prefetch(rowp + 64, 0, 1);
    const float4* row = (const float4*)rowp;         // rows are 320B, 16B aligned
    float m = -INFINITY;
#pragma unroll
    for (int i = 0; i < L_ / 4; ++i) {
        float4 v = row[i];
        m = fmaxf(m, fmaxf(fmaxf(v.x, v.y), fmaxf(v.z, v.w)));
    }
    keys[r] = f2ord(m);
}

// ---------------------------------------------------------------------------
// Kernel 2: per-batch exact top-300 with JAX ordering (desc value, asc index
// tie-break). One 1024-thread block per batch; keys live in LDS.
// ---------------------------------------------------------------------------
__global__ __launch_bounds__(1024)
void topk_kernel(const unsigned* __restrict__ keys_g, int* __restrict__ topk) {
    __shared__ unsigned keys[A_];        // 33.6 KB
    __shared__ unsigned hist[256];
    __shared__ unsigned scanbuf[1024];
    __shared__ unsigned cand_key[K_];
    __shared__ int      cand_idx[K_];
    __shared__ unsigned sh_prefix, sh_kremain, sh_cnt;

    const int b = blockIdx.x;
    const int t = threadIdx.x;
    const unsigned* kg = keys_g + (size_t)b * A_;

    for (int i = t; i < A_; i += 1024) keys[i] = kg[i];
    if (t == 0) { sh_prefix = 0u; sh_kremain = K_; sh_cnt = 0u; }
    __syncthreads();

    // 4-pass MSB radix select: find exact 300th-largest key value.
    unsigned prefMask = 0u;
    for (int shift = 24; shift >= 0; shift -= 8) {
        if (t < 256) hist[t] = 0u;
        __syncthreads();
        unsigned prefix = sh_prefix;
        for (int i = t; i < A_; i += 1024) {
            unsigned k = keys[i];
            if ((k & prefMask) == prefix)
                atomicAdd(&hist[(k >> shift) & 255u], 1u);
        }
        __syncthreads();
        if (t == 0) {
            unsigned kr = sh_kremain;
            unsigned c = 0u;
            int sel = 0;
            for (int bin = 255; bin >= 0; --bin) {
                unsigned nc = c + hist[bin];
                if (nc >= kr) { sel = bin; sh_kremain = kr - c; break; }
                c = nc;
            }
            sh_prefix = prefix | ((unsigned)sel << shift);
        }
        __syncthreads();
        prefMask |= (255u << shift);
    }

    const unsigned T   = sh_prefix;        // 300th-largest key
    const int     need = (int)sh_kremain;  // how many == T to take
    const int     cgr  = K_ - need;        // count of keys strictly > T

    // Strictly-greater candidates (order fixed later by rank sort).
    for (int i = t; i < A_; i += 1024) {
        unsigned k = keys[i];
        if (k > T) {
            unsigned p = atomicAdd(&sh_cnt, 1u);
            cand_key[p] = k; cand_idx[p] = i;
        }
    }
    __syncthreads();

    // Equal-to-T candidates in ascending-index order (stable tie-break):
    // chunked count + block-wide inclusive scan + rank assignment.
    const int CH = (A_ + 1023) / 1024;     // 9 rows per thread
    int lo = t * CH;
    int hi = lo + CH; if (hi > A_) hi = A_;
    if (lo > A_) lo = A_;
    unsigned ceq = 0;
    for (int i = lo; i < hi; ++i) if (keys[i] == T) ++ceq;
    scanbuf[t] = ceq;
    __syncthreads();
    for (int off = 1; off < 1024; off <<= 1) {
        unsigned v = (t >= off) ? scanbuf[t - off] : 0u;
        __syncthreads();
        scanbuf[t] += v;
        __syncthreads();
    }
    int rank = (int)(scanbuf[t] - ceq);    // exclusive prefix for my chunk
    for (int i = lo; i < hi; ++i) {
        if (keys[i] == T) {
            if (rank < need) { cand_key[cgr + rank] = T; cand_idx[cgr + rank] = i; }
            ++rank;
        }
    }
    __syncthreads();

    // Rank-sort the 300 survivors: descending key, ascending index.
    if (t < K_) {
        unsigned mk = cand_key[t];
        int      mi = cand_idx[t];
        int r = 0;
        for (int j = 0; j < K_; ++j) {
            unsigned k = cand_key[j];
            int     ji = cand_idx[j];
            if (k > mk || (k == mk && ji < mi)) ++r;
        }
        topk[(size_t)b * K_ + r] = mi;
    }
}

// ---------------------------------------------------------------------------
// Kernel 3: gather top-300 rows + sigmoid. One 128-thread block per (b,q).
// The 1KB memory row goes through the CDNA5 async global->LDS path.
// ---------------------------------------------------------------------------
__global__ __launch_bounds__(128)
void gather_kernel(const float* __restrict__ cls,
                   const float* __restrict__ coord,
                   const float* __restrict__ mem,
                   const int* __restrict__ topk,
                   float* __restrict__ out) {
    __shared__ float stage[H_];            // 1KB staging for target row
    const int blk = blockIdx.x;            // = b*K + q
    const int b = blk / K_;
    const int t = threadIdx.x;
    const int idx = topk[blk];

    float* out0 = out;                           // init_reference_points (B,K,4)
    float* out1 = out  + (size_t)B_ * K_ * 4;    // target               (B,K,H)
    float* out2 = out1 + (size_t)B_ * K_ * H_;   // enc_topk_logits      (B,K,L)
    float* out3 = out2 + (size_t)B_ * K_ * L_;   // enc_topk_bboxes      (B,K,4)

    const float* mrow = mem + ((size_t)b * A_ + idx) * H_;

    // Async copy: 64 lanes x 16B global -> LDS (ASYNCcnt-tracked).
    if (t < H_ / 4) {
        unsigned long long ga = (unsigned long long)(uintptr_t)(const void*)(mrow + t * 4);
        unsigned ldso = (unsigned)(uintptr_t)(const void*)&stage[t * 4];
        asm volatile("global_load_async_to_lds_b128 %0, %1, off"
                     :: "v"(ldso), "v"(ga)
                     : "memory");
    }
#if __has_builtin(__builtin_amdgcn_s_wait_asynccnt)
    __builtin_amdgcn_s_wait_asynccnt(0);
#else
    asm volatile("s_wait_asynccnt 0" ::: "memory");
#endif
    __syncthreads();

    if (t < H_ / 4) {
        float4 v = ((const float4*)stage)[t];
        ((float4*)(out1 + (size_t)blk * H_))[t] = v;
    }
    if (t < L_) {
        out2[(size_t)blk * L_ + t] = cls[((size_t)b * A_ + idx) * L_ + t];
    }
    if (t < 4) {
        float c = coord[((size_t)b * A_ + idx) * 4 + t];
        out0[(size_t)blk * 4 + t] = c;
        // numerically stable sigmoid
        float s = (c >= 0.0f) ? (1.0f / (1.0f + expf(-c)))
                              : (expf(c) / (1.0f + expf(c)));
        out3[(size_t)blk * 4 + t] = s;
    }
}

// ---------------------------------------------------------------------------
extern "C" void kernel_launch(void* const* d_in, const int* in_sizes, int n_in,
                              void* d_out, int out_size, void* d_ws, size_t ws_size,
                              hipStream_t stream) {
    const float* cls   = (const float*)d_in[0];  // (B, A, L)
    const float* coord = (const float*)d_in[1];  // (B, A, 4)
    const float* mem   = (const float*)d_in[2];  // (B, A, H)
    // d_in[3] (sources_last_element) is unused by the reference output.

    unsigned* keys = (unsigned*)d_ws;                                  // B*A u32
    int* topk = (int*)((char*)d_ws + (size_t)B_ * A_ * sizeof(unsigned)); // B*K i32
    float* out = (float*)d_out;

    rowmax_kernel<<<(B_ * A_ + 255) / 256, 256, 0, stream>>>(cls, keys);
    topk_kernel<<<B_, 1024, 0, stream>>>(keys, topk);
    gather_kernel<<<B_ * K_, 128, 0, stream>>>(cls, coord, mem, topk, out);
}